// GraphGCN_13718125543732
// MI455X (gfx1250) — compile-verified
//
#include <hip/hip_runtime.h>
#include <hip/hip_bf16.h>

typedef __attribute__((ext_vector_type(2))) float v2f;
typedef __attribute__((ext_vector_type(8))) float v8f;

#define D_FEAT 64

// ---------------------------------------------------------------------------
// Kernel 1: zero accumulators (buf1 = x1-sum, deg, d_out = x2-sum)
// ---------------------------------------------------------------------------
__global__ void gcn_zero(float* __restrict__ buf1, float* __restrict__ deg,
                         float* __restrict__ dout, int n_nodes) {
  const int nfeat = n_nodes * D_FEAT;
  const int total = 2 * nfeat + n_nodes;
  for (int i = blockIdx.x * blockDim.x + threadIdx.x; i < total;
       i += gridDim.x * blockDim.x) {
    if (i < nfeat)            buf1[i] = 0.0f;
    else if (i < 2 * nfeat)   dout[i - nfeat] = 0.0f;
    else                      deg[i - 2 * nfeat] = 0.0f;
  }
}

// ---------------------------------------------------------------------------
// Kernel 2/4: edge-parallel scatter-add. One wave32 per edge; each lane moves
// a float2 of the 64-float row (coalesced 256B) and issues two
// global_atomic_add_f32 (L2-resident on MI455X: whole table fits in 192MB L2).
// deg != nullptr only on round 1.
// ---------------------------------------------------------------------------
__global__ void gcn_scatter(const float* __restrict__ h,
                            const int* __restrict__ src,
                            const int* __restrict__ dst,
                            float* __restrict__ accum,
                            float* __restrict__ deg,
                            int n_edges, int n_nodes) {
  const int lane = threadIdx.x & 31;
  const int edge = (blockIdx.x * blockDim.x + threadIdx.x) >> 5;
  if (edge >= n_edges) return;
  const int s = src[edge];
  const int d = dst[edge];
  if ((unsigned)s >= (unsigned)n_nodes || (unsigned)d >= (unsigned)n_nodes) return;
  const v2f v = *(const v2f*)(h + s * D_FEAT + lane * 2);
  float* ap = accum + d * D_FEAT + lane * 2;
  unsafeAtomicAdd(ap + 0, v.x);
  unsafeAtomicAdd(ap + 1, v.y);
  if (deg != nullptr && lane == 0) unsafeAtomicAdd(deg + d, 1.0f);
}

// ---------------------------------------------------------------------------
// Kernel 3: normalize in place: buf = deg>0 ? buf/deg : 0
// ---------------------------------------------------------------------------
__global__ void gcn_norm(float* __restrict__ buf, const float* __restrict__ deg,
                         int n_nodes) {
  const int i = blockIdx.x * blockDim.x + threadIdx.x;
  if (i >= n_nodes * D_FEAT) return;
  const float dg = deg[i >> 6];
  buf[i] = (dg > 0.0f) ? buf[i] / dg : 0.0f;
}

// ---------------------------------------------------------------------------
// Kernel 5: WMMA finalize. One wave per 16-node tile. For the tile:
//   G12 = X1*X2^T, G11 = X1*X1^T, G22 = X2*X2^T via v_wmma_f32_16x16x4_f32
//   (K=64 -> 16 chained WMMAs each; diagonals give dot, |x1|^2, |x2|^2).
// f32 16x16x4 layout property: the per-lane float2 at
//   X[tile*16 + lane%16][k0 + 2*(lane/16)]
// is simultaneously the correct A-operand (16x4 of X) and the correct
// B-operand (4x16 of X^T), so one load serves both sides.
// Diagonal of C/D: node v -> lane v, VGPR v; node 8+v -> lane 24+v, VGPR v.
// Then blend out = w*x2 + (1-w)*x1 in place over d_out (which held x2-sum).
// blockDim = 64 (2 waves) so every wave owns a full tile: EXEC is all ones
// around the WMMAs as required by the ISA.
// ---------------------------------------------------------------------------
__global__ void gcn_finalize(const float* __restrict__ x1,
                             float* __restrict__ out,        // in: x2-sum, out: result
                             const float* __restrict__ deg) {
  __shared__ float w_s[2][16];
  __shared__ float iv_s[2][16];
  const int lane = threadIdx.x & 31;
  const int wv   = threadIdx.x >> 5;
  const int tile = blockIdx.x * 2 + wv;

  const int r     = lane & 15;       // node-in-tile for A rows / B cols
  const int khalf = lane >> 4;       // which K pair this lane holds
  const int node  = tile * 16 + r;

  const float dg = deg[node];
  const float iv = (dg > 0.0f) ? 1.0f / dg : 0.0f;   // x2 = x2sum * iv

  const float* x1p = x1  + node * D_FEAT;
  const float* x2p = out + node * D_FEAT;

  v8f c12 = {}; v8f c11 = {}; v8f c22 = {};
#pragma unroll
  for (int kc = 0; kc < 16; ++kc) {
    const int col = kc * 4 + khalf * 2;
    const v2f a  = *(const v2f*)(x1p + col);
    const v2f bs = *(const v2f*)(x2p + col);
    v2f b; b.x = bs.x * iv; b.y = bs.y * iv;
    c12 = __builtin_amdgcn_wmma_f32_16x16x4_f32(false, a, false, b, (short)0, c12, false, false);
    c11 = __builtin_amdgcn_wmma_f32_16x16x4_f32(false, a, false, a, (short)0, c11, false, false);
    c22 = __builtin_amdgcn_wmma_f32_16x16x4_f32(false, b, false, b, (short)0, c22, false, false);
  }

  if (lane < 16) iv_s[wv][lane] = iv;

  // Pull the Gram diagonals: all three live at the same (lane, vgpr) slot.
  float dt = 0.0f, s11 = 0.0f, s22 = 0.0f;
  int myi = -1;
#pragma unroll
  for (int v = 0; v < 8; ++v) {
    if (lane == v)      { dt = c12[v]; s11 = c11[v]; s22 = c22[v]; myi = v; }
    if (lane == 24 + v) { dt = c12[v]; s11 = c11[v]; s22 = c22[v]; myi = 8 + v; }
  }
  if (myi >= 0) {
    const float w = dt / fmaxf(sqrtf(s11) * sqrtf(s22), 1e-8f);
    w_s[wv][myi] = w;
  }
  __syncthreads();

  // Blend 16 nodes x 64 feats = 1024 contiguous floats starting at tile*1024.
  const int base = tile * 16 * D_FEAT;
  for (int t = lane; t < 16 * D_FEAT; t += 32) {
    const int i  = t >> 6;
    const float w  = w_s[wv][i];
    const float ivv = iv_s[wv][i];
    const float a = x1[base + t];
    const float b = out[base + t] * ivv;
    out[base + t] = w * b + (1.0f - w) * a;
  }
}

// ---------------------------------------------------------------------------
extern "C" void kernel_launch(void* const* d_in, const int* in_sizes, int n_in,
                              void* d_out, int out_size, void* d_ws, size_t ws_size,
                              hipStream_t stream) {
  const float* features = (const float*)d_in[0];
  const int*   edges    = (const int*)d_in[1];

  const int n_nodes = in_sizes[0] / D_FEAT;   // 100000
  const int n_edges = in_sizes[1] / 2;        // 1600000
  const int*   src  = edges;
  const int*   dst  = edges + n_edges;

  float* buf1 = (float*)d_ws;                       // x1 sum -> x1 (25.6 MB)
  float* degp = buf1 + (size_t)n_nodes * D_FEAT;    // degree  (0.4 MB)
  float* outp = (float*)d_out;                      // x2 sum -> result

  // 1) zero accumulators
  gcn_zero<<<4096, 256, 0, stream>>>(buf1, degp, outp, n_nodes);

  // 2) round-1 scatter: features[src] += into buf1, count deg
  const int scatter_blocks = (n_edges * 32 + 255) / 256;
  gcn_scatter<<<scatter_blocks, 256, 0, stream>>>(features, src, dst, buf1, degp,
                                                  n_edges, n_nodes);

  // 3) x1 = buf1 / deg (in place)
  gcn_norm<<<(n_nodes * D_FEAT + 255) / 256, 256, 0, stream>>>(buf1, degp, n_nodes);

  // 4) round-2 scatter: x1[src] += into d_out (x2 sum); deg unchanged
  gcn_scatter<<<scatter_blocks, 256, 0, stream>>>(buf1, src, dst, outp, nullptr,
                                                  n_edges, n_nodes);

  // 5) cosine blend via f32 WMMA Gram diagonals (n_nodes is a multiple of 16)
  const int ntiles = n_nodes / 16;                  // 6250
  gcn_finalize<<<ntiles / 2, 64, 0, stream>>>(buf1, outp, degp);
}